// ImagBehavior_67284957659345
// MI455X (gfx1250) — compile-verified
//
#include <hip/hip_runtime.h>

// lambda-return reverse scan, T=16, B=524288.
// Memory-bound streaming kernel: ~126 MiB of one-touch traffic, ~24 MFLOP.
// DRAM floor on MI455X (23.3 TB/s) ~= 5.5 us. Strategy: b128 NT loads/stores,
// 4 columns per thread, full unroll over the 15 time steps, global_prefetch
// of the next time-row (gfx1250 global_prefetch_b8 path).

typedef float v4f __attribute__((ext_vector_type(4)));

#define LAM  0.95f
#define OML  0.05f   // 1 - LAM

template <int T>
__global__ __launch_bounds__(256) void lambda_return_v4(
    const float* __restrict__ reward,
    const float* __restrict__ value,
    const float* __restrict__ discount,
    float* __restrict__ out,
    int B)
{
    const long c = (long)blockIdx.x * blockDim.x + threadIdx.x; // vec4 column
    const long b = c * 4;
    if (b >= B) return;

    // bootstrap: acc = value[T-1]
    v4f acc = __builtin_nontemporal_load(
        (const v4f*)(value + (long)(T - 1) * B + b));

#pragma unroll
    for (int i = T - 2; i >= 0; --i) {
        // Prefetch the next (earlier) time-row while this step's loads/FMAs run.
        if (i > 0) {
            __builtin_prefetch(reward   + (long)(i - 1) * B + b, 0, 1);
            __builtin_prefetch(discount + (long)(i - 1) * B + b, 0, 1);
            __builtin_prefetch(value    + (long)i       * B + b, 0, 1);
        }
        v4f r = __builtin_nontemporal_load((const v4f*)(reward   + (long)i * B + b));
        v4f d = __builtin_nontemporal_load((const v4f*)(discount + (long)i * B + b));
        v4f v = __builtin_nontemporal_load((const v4f*)(value + (long)(i + 1) * B + b));

        v4f t;
#pragma unroll
        for (int k = 0; k < 4; ++k)
            t[k] = __builtin_fmaf(LAM, acc[k], OML * v[k]);
#pragma unroll
        for (int k = 0; k < 4; ++k)
            acc[k] = __builtin_fmaf(d[k], t[k], r[k]);

        __builtin_nontemporal_store(acc, (v4f*)(out + (long)i * B + b));
    }
}

// Scalar fallback for B not divisible by 4 (not hit for the reference shapes).
template <int T>
__global__ __launch_bounds__(256) void lambda_return_s1(
    const float* __restrict__ reward,
    const float* __restrict__ value,
    const float* __restrict__ discount,
    float* __restrict__ out,
    int B)
{
    const long b = (long)blockIdx.x * blockDim.x + threadIdx.x;
    if (b >= B) return;
    float acc = value[(long)(T - 1) * B + b];
#pragma unroll
    for (int i = T - 2; i >= 0; --i) {
        float r = reward[(long)i * B + b];
        float d = discount[(long)i * B + b];
        float v = value[(long)(i + 1) * B + b];
        float t = __builtin_fmaf(LAM, acc, OML * v);
        acc = __builtin_fmaf(d, t, r);
        out[(long)i * B + b] = acc;
    }
}

extern "C" void kernel_launch(void* const* d_in, const int* in_sizes, int n_in,
                              void* d_out, int out_size, void* d_ws, size_t ws_size,
                              hipStream_t stream) {
    (void)n_in; (void)out_size; (void)d_ws; (void)ws_size;
    const float* reward   = (const float*)d_in[0];
    const float* value    = (const float*)d_in[1];
    const float* discount = (const float*)d_in[2];
    float* out = (float*)d_out;

    constexpr int T = 16;                 // fixed by the reference setup
    const int B = in_sizes[0] / T;        // 524288

    if ((B & 3) == 0) {
        const int vcols = B / 4;
        dim3 block(256);
        dim3 grid((vcols + 255) / 256);
        lambda_return_v4<T><<<grid, block, 0, stream>>>(reward, value, discount, out, B);
    } else {
        dim3 block(256);
        dim3 grid((B + 255) / 256);
        lambda_return_s1<T><<<grid, block, 0, stream>>>(reward, value, discount, out, B);
    }
}